// LAME_2568390443509
// MI455X (gfx1250) — compile-verified
//
#include <hip/hip_runtime.h>
#include <math.h>

// ---------------- problem constants (match reference) ----------------
#define N_PTS   4096
#define D_FEAT  2048
#define N_CLS   256
#define KNN_K   5
#define MAX_STEPS 100

typedef float v2f __attribute__((ext_vector_type(2)));
typedef float v8f __attribute__((ext_vector_type(8)));

// V_WMMA_F32_16X16X4_F32 : D(16x16 f32) = A(16x4 f32) * B(4x16 f32) + C
static __device__ __forceinline__ v8f wmma_f32_16x16x4(v2f a, v2f b, v8f c) {
  return __builtin_amdgcn_wmma_f32_16x16x4_f32(
      /*neg_a=*/false, a, /*neg_b=*/false, b,
      /*c_mod=*/(short)0, c, /*reuse_a=*/false, /*reuse_b=*/false);
}

// ---------------------------------------------------------------------
// Row L2-norm reciprocal of x [N_PTS x D_FEAT]
// ---------------------------------------------------------------------
__global__ void rownorm_kernel(const float* __restrict__ x, float* __restrict__ rn) {
  __shared__ float red[256];
  const int row = blockIdx.x, t = threadIdx.x;
  const float* xr = x + (size_t)row * D_FEAT;
  float s = 0.f;
  for (int j = t; j < D_FEAT; j += 256) { float v = xr[j]; s += v * v; }
  red[t] = s; __syncthreads();
  for (int st = 128; st > 0; st >>= 1) { if (t < st) red[t] += red[t + st]; __syncthreads(); }
  if (t == 0) rn[row] = rsqrtf(fmaxf(red[0], 1e-30f));
}

// ---------------------------------------------------------------------
// C[M,N] = rowScale .* (A[M,K] * B[N,K]^T) .* colScale + bias
// Wave computes a 64x64 tile (4x4 WMMA subtiles); block = 4 waves stacked
// in M (block tile 256 x 64). grid = (N/64, M/256), blockDim = 128.
// K, N compile-time so fragment addresses are base-ptr + constant offset.
// ---------------------------------------------------------------------
template <int K, int N>
__global__ __launch_bounds__(128, 1)
void gemm_abt64(const float* __restrict__ A, const float* __restrict__ B,
                float* __restrict__ C,
                const float* __restrict__ rowScale,
                const float* __restrict__ colScale,
                const float* __restrict__ bias) {
  const int lane = threadIdx.x & 31;
  const int wave = threadIdx.x >> 5;
  const int half = lane >> 4;       // selects K-pair {0,1} vs {2,3}
  const int l    = lane & 15;       // M (A) / N (B) index within 16-tile

  const int row0 = (blockIdx.y * 4 + wave) * 64;
  const int col0 = blockIdx.x * 64;

  // one base pointer per matrix; all tile offsets are compile-time immediates
  const float* pa = A + (size_t)(row0 + l) * K + 2 * half;
  const float* pb = B + (size_t)(col0 + l) * K + 2 * half;

  v8f acc[4][4];
  #pragma unroll
  for (int i = 0; i < 4; i++)
    #pragma unroll
    for (int j = 0; j < 4; j++) acc[i][j] = {};

  for (int k = 0; k < K; k += 16) {
    #pragma unroll
    for (int kk = 0; kk < 4; kk++) {
      v2f a[4], b[4];
      #pragma unroll
      for (int i = 0; i < 4; i++)
        a[i] = *(const v2f*)(pa + i * 16 * K + kk * 4);
      #pragma unroll
      for (int j = 0; j < 4; j++)
        b[j] = *(const v2f*)(pb + j * 16 * K + kk * 4);
      #pragma unroll
      for (int i = 0; i < 4; i++)
        #pragma unroll
        for (int j = 0; j < 4; j++)
          acc[i][j] = wmma_f32_16x16x4(a[i], b[j], acc[i][j]);
    }
    pa += 16;
    pb += 16;
  }

  float* pc = C + (size_t)(row0 + 8 * half) * N + col0 + l;
  #pragma unroll
  for (int i = 0; i < 4; i++) {
    #pragma unroll
    for (int r = 0; r < 8; r++) {
      const int row = row0 + 8 * half + i * 16 + r;
      const float rs = rowScale ? rowScale[row] : 1.0f;
      #pragma unroll
      for (int j = 0; j < 4; j++) {
        float v = acc[i][j][r] * rs;
        if (colScale) v *= colScale[col0 + l + 16 * j];
        if (bias)     v += bias[col0 + l + 16 * j];
        pc[(i * 16 + r) * N + j * 16] = v;
      }
    }
  }
}

// ---------------------------------------------------------------------
// C[M,N] = A[M,K] * B[K,N]   (B row-major [K,N]) — used for Kmat @ Y
// Same 64x64-per-wave tiling. grid = (N/64, M/256), blockDim = 128.
// ---------------------------------------------------------------------
template <int K, int N>
__global__ __launch_bounds__(128, 1)
void gemm_ab64(const float* __restrict__ A, const float* __restrict__ B,
               float* __restrict__ C) {
  const int lane = threadIdx.x & 31;
  const int wave = threadIdx.x >> 5;
  const int half = lane >> 4;
  const int l    = lane & 15;

  const int row0 = (blockIdx.y * 4 + wave) * 64;
  const int col0 = blockIdx.x * 64;

  const float* pa = A + (size_t)(row0 + l) * K + 2 * half;
  const float* pb = B + (size_t)(2 * half) * N + col0 + l;

  v8f acc[4][4];
  #pragma unroll
  for (int i = 0; i < 4; i++)
    #pragma unroll
    for (int j = 0; j < 4; j++) acc[i][j] = {};

  for (int k = 0; k < K; k += 16) {
    #pragma unroll
    for (int kk = 0; kk < 4; kk++) {
      v2f a[4], b[4];
      #pragma unroll
      for (int i = 0; i < 4; i++)
        a[i] = *(const v2f*)(pa + i * 16 * K + kk * 4);
      #pragma unroll
      for (int j = 0; j < 4; j++) {
        b[j].x = pb[kk * 4 * N + j * 16];
        b[j].y = pb[kk * 4 * N + N + j * 16];
      }
      #pragma unroll
      for (int i = 0; i < 4; i++)
        #pragma unroll
        for (int j = 0; j < 4; j++)
          acc[i][j] = wmma_f32_16x16x4(a[i], b[j], acc[i][j]);
    }
    pa += 16;
    pb += 16 * N;
  }

  float* pc = C + (size_t)(row0 + 8 * half) * N + col0 + l;
  #pragma unroll
  for (int i = 0; i < 4; i++)
    #pragma unroll
    for (int r = 0; r < 8; r++)
      #pragma unroll
      for (int j = 0; j < 4; j++)
        pc[(i * 16 + r) * N + j * 16] = acc[i][j][r];
}

// ---------------------------------------------------------------------
// In-place: logits -> unary = -log(softmax(logits)+1e-10); Y0 = softmax(-unary)
// One block per row, 256 threads == N_CLS.
// ---------------------------------------------------------------------
__global__ void unary_y0_kernel(float* __restrict__ UL, float* __restrict__ Y0) {
  __shared__ float red[256];
  const int row = blockIdx.x, t = threadIdx.x;
  const size_t idx = (size_t)row * N_CLS + t;
  const float logit = UL[idx];

  red[t] = logit; __syncthreads();
  for (int s = 128; s > 0; s >>= 1) { if (t < s) red[t] = fmaxf(red[t], red[t + s]); __syncthreads(); }
  const float m = red[0]; __syncthreads();
  const float e = expf(logit - m);
  red[t] = e; __syncthreads();
  for (int s = 128; s > 0; s >>= 1) { if (t < s) red[t] += red[t + s]; __syncthreads(); }
  const float p = e / red[0]; __syncthreads();
  const float u = -logf(p + 1e-10f);
  UL[idx] = u;

  const float nu = -u;
  red[t] = nu; __syncthreads();
  for (int s = 128; s > 0; s >>= 1) { if (t < s) red[t] = fmaxf(red[t], red[t + s]); __syncthreads(); }
  const float m2 = red[0]; __syncthreads();
  const float e2 = expf(nu - m2);
  red[t] = e2; __syncthreads();
  for (int s = 128; s > 0; s >>= 1) { if (t < s) red[t] += red[t + s]; __syncthreads(); }
  Y0[idx] = e2 / red[0];
}

// ---------------------------------------------------------------------
// Per row of normalized gram G: dist_j = sqrt(max(2-2g,0)); 5th smallest.
// ---------------------------------------------------------------------
__global__ void kth_dist_kernel(const float* __restrict__ G, float* __restrict__ kth) {
  __shared__ float lists[256 * KNN_K];
  const int row = blockIdx.x, t = threadIdx.x;
  const float* g = G + (size_t)row * N_PTS;

  float best[KNN_K];
  #pragma unroll
  for (int i = 0; i < KNN_K; i++) best[i] = INFINITY;

  for (int j = t; j < N_PTS; j += 256) {
    const float d2 = fmaxf(2.0f - 2.0f * g[j], 0.0f);
    const float d = sqrtf(d2);
    if (d < best[KNN_K - 1]) {
      best[KNN_K - 1] = d;
      #pragma unroll
      for (int k = KNN_K - 1; k > 0; k--)
        if (best[k] < best[k - 1]) { float tmp = best[k]; best[k] = best[k - 1]; best[k - 1] = tmp; }
    }
  }
  #pragma unroll
  for (int i = 0; i < KNN_K; i++) lists[t * KNN_K + i] = best[i];
  __syncthreads();

  for (int s = 128; s > 0; s >>= 1) {
    if (t < s) {
      float* la = &lists[t * KNN_K];
      float* lb = &lists[(t + s) * KNN_K];
      float out[KNN_K]; int a = 0, b = 0;
      #pragma unroll
      for (int m = 0; m < KNN_K; m++) out[m] = (la[a] <= lb[b]) ? la[a++] : lb[b++];
      #pragma unroll
      for (int m = 0; m < KNN_K; m++) la[m] = out[m];
    }
    __syncthreads();
  }
  if (t == 0) kth[row] = lists[KNN_K - 1];
}

// sigma = mean(kth); scal[0] = 1/(2 sigma^2)
__global__ void sigma_kernel(const float* __restrict__ kth, float* __restrict__ scal) {
  __shared__ float red[256];
  const int t = threadIdx.x;
  float s = 0.f;
  for (int j = t; j < N_PTS; j += 256) s += kth[j];
  red[t] = s; __syncthreads();
  for (int st = 128; st > 0; st >>= 1) { if (t < st) red[t] += red[t + st]; __syncthreads(); }
  if (t == 0) {
    const float sigma = red[0] / (float)N_PTS;
    scal[0] = 1.0f / (2.0f * sigma * sigma);
  }
}

// G <- exp(-max(2-2g,0) * inv2sig), elementwise, float4 per thread
__global__ void kmat_kernel(float* __restrict__ G, const float* __restrict__ scal) {
  const float inv = scal[0];
  const size_t i = ((size_t)blockIdx.x * blockDim.x + threadIdx.x) * 4;
  float4 g = *(float4*)(G + i);
  g.x = expf(-fmaxf(2.0f - 2.0f * g.x, 0.0f) * inv);
  g.y = expf(-fmaxf(2.0f - 2.0f * g.y, 0.0f) * inv);
  g.z = expf(-fmaxf(2.0f - 2.0f * g.z, 0.0f) * inv);
  g.w = expf(-fmaxf(2.0f - 2.0f * g.w, 0.0f) * inv);
  *(float4*)(G + i) = g;
}

__global__ void init_state_kernel(float* __restrict__ scal, int* __restrict__ flags) {
  if (threadIdx.x == 0) { scal[1] = INFINITY; flags[0] = 0; }
}

// One bound-update step (rowwise part): Ynew = softmax(pairwise - unary);
// per-row energy partial. Freezes state once flags[0] (done) is set.
__global__ void step_row_kernel(const float* __restrict__ unary, const float* __restrict__ P,
                                const float* __restrict__ Ycur, float* __restrict__ Ynext,
                                float* __restrict__ Epart, const int* __restrict__ flags) {
  __shared__ float red[256];
  const int row = blockIdx.x, t = threadIdx.x;
  const size_t idx = (size_t)row * N_CLS + t;
  if (flags[0]) {                         // uniform branch: converged -> freeze
    Ynext[idx] = Ycur[idx];
    if (t == 0) Epart[row] = 0.0f;
    return;
  }
  const float u  = unary[idx];
  const float pw = P[idx];                // = BOUND_LAMBDA * (K@Y), lambda = 1
  const float lg = pw - u;

  red[t] = lg; __syncthreads();
  for (int s = 128; s > 0; s >>= 1) { if (t < s) red[t] = fmaxf(red[t], red[t + s]); __syncthreads(); }
  const float m = red[0]; __syncthreads();
  const float e = expf(lg - m);
  red[t] = e; __syncthreads();
  for (int s = 128; s > 0; s >>= 1) { if (t < s) red[t] += red[t + s]; __syncthreads(); }
  const float y = e / red[0]; __syncthreads();

  const float ee = u * y - pw * y + y * logf(fmaxf(y, 1e-20f));
  red[t] = ee; __syncthreads();
  for (int s = 128; s > 0; s >>= 1) { if (t < s) red[t] += red[t + s]; __syncthreads(); }
  if (t == 0) Epart[row] = red[0];
  Ynext[idx] = y;
}

// Reduce energy, update convergence state (matches reference scan-step order).
__global__ void update_kernel(const float* __restrict__ Epart, float* __restrict__ scal,
                              int* __restrict__ flags, int stepIdx) {
  __shared__ float red[256];
  const int t = threadIdx.x;
  float s = 0.f;
  for (int j = t; j < N_PTS; j += 256) s += Epart[j];
  red[t] = s; __syncthreads();
  for (int st = 128; st > 0; st >>= 1) { if (t < st) red[t] += red[t + st]; __syncthreads(); }
  if (t == 0) {
    const float E = red[0];
    const int   done = flags[0];
    const float oldE = scal[1];
    const int conv = (stepIdx > 1) && (fabsf(E - oldE) <= 1e-8f * fabsf(oldE));
    scal[1]  = done ? oldE : E;   // Ek = where(done, oldE, E)
    flags[0] = done | conv;
  }
}

// ---------------------------------------------------------------------
extern "C" void kernel_launch(void* const* d_in, const int* in_sizes, int n_in,
                              void* d_out, int out_size, void* d_ws, size_t ws_size,
                              hipStream_t stream) {
  (void)in_sizes; (void)n_in; (void)out_size; (void)ws_size;
  const float* x    = (const float*)d_in[0];   // [4096, 2048]
  const float* W    = (const float*)d_in[1];   // [256, 2048]
  const float* bias = (const float*)d_in[2];   // [256]

  char* base = (char*)d_ws;
  size_t off = 0;
  auto alloc = [&](size_t bytes) -> void* {
    void* p = base + off;
    off = (off + bytes + 255) & ~(size_t)255;
    return p;
  };
  float* G     = (float*)alloc((size_t)N_PTS * N_PTS * sizeof(float)); // 64 MB, gram -> Kmat
  float* U     = (float*)alloc((size_t)N_PTS * N_CLS * sizeof(float)); // logits -> unary
  float* Ya    = (float*)alloc((size_t)N_PTS * N_CLS * sizeof(float));
  float* Yb    = (float*)alloc((size_t)N_PTS * N_CLS * sizeof(float));
  float* P     = (float*)alloc((size_t)N_PTS * N_CLS * sizeof(float));
  float* rn    = (float*)alloc(N_PTS * sizeof(float));
  float* kth   = (float*)alloc(N_PTS * sizeof(float));
  float* Epart = (float*)alloc(N_PTS * sizeof(float));
  float* scal  = (float*)alloc(256);   // [0]=inv2sig, [1]=oldE
  int*   flags = (int*)alloc(256);     // [0]=done

  // 1) reciprocal row norms of x
  rownorm_kernel<<<N_PTS, 256, 0, stream>>>(x, rn);

  // 2) logits = x @ W^T + b   (WMMA f32, A.B^T form)
  gemm_abt64<D_FEAT, N_CLS><<<dim3(N_CLS / 64, N_PTS / 256), 128, 0, stream>>>(
      x, W, U, nullptr, nullptr, bias);

  // 3) unary + Y0
  unary_y0_kernel<<<N_PTS, 256, 0, stream>>>(U, Ya);

  // 4) normalized gram: G = diag(rn) (x x^T) diag(rn)  (WMMA f32)
  gemm_abt64<D_FEAT, N_PTS><<<dim3(N_PTS / 64, N_PTS / 256), 128, 0, stream>>>(
      x, x, G, rn, rn, nullptr);

  // 5) 5th-smallest distance per row; 6) sigma; 7) G -> RBF kernel in place
  kth_dist_kernel<<<N_PTS, 256, 0, stream>>>(G, kth);
  sigma_kernel<<<1, 256, 0, stream>>>(kth, scal);
  kmat_kernel<<<(int)(((size_t)N_PTS * N_PTS) / 4 / 256), 256, 0, stream>>>(G, scal);

  // 8) convergence state
  init_state_kernel<<<1, 32, 0, stream>>>(scal, flags);

  // 9) 100 bound-update steps (dominant cost: K@Y WMMA GEMM each step)
  float* cur = Ya;
  float* nxt = Yb;
  for (int i = 0; i < MAX_STEPS; i++) {
    gemm_ab64<N_PTS, N_CLS><<<dim3(N_CLS / 64, N_PTS / 256), 128, 0, stream>>>(G, cur, P);
    step_row_kernel<<<N_PTS, 256, 0, stream>>>(U, P, cur, nxt, Epart, flags);
    update_kernel<<<1, 256, 0, stream>>>(Epart, scal, flags, i);
    float* tmp = cur; cur = nxt; nxt = tmp;
  }

  hipMemcpyAsync(d_out, cur, (size_t)N_PTS * N_CLS * sizeof(float),
                 hipMemcpyDeviceToDevice, stream);
}